// AttentionADCell_21852793602283
// MI455X (gfx1250) — compile-verified
//
#include <hip/hip_runtime.h>
#include <hip/hip_bf16.h>
#include <math.h>

typedef __attribute__((ext_vector_type(2))) float v2f;
typedef __attribute__((ext_vector_type(8))) float v8f;

#define EPSF 1e-6f

// gfx1250 hardware tanh (new TRANS op on CDNA5). Prefer the builtin; fall
// back to inline asm (v_nop satisfies the 1-op TRANS result hazard window).
static __device__ inline float fast_tanh(float x) {
#if __has_builtin(__builtin_amdgcn_tanhf)
  return __builtin_amdgcn_tanhf(x);
#else
  float r;
  asm volatile("v_tanh_f32 %0, %1\n\tv_nop" : "=v"(r) : "v"(x));
  return r;
#endif
}

// ---------------------------------------------------------------------------
// Zero a float buffer (q region must be re-zeroed every call: atomics below).
// ---------------------------------------------------------------------------
__global__ void k_zero(float* __restrict__ p, int n) {
  int i = blockIdx.x * blockDim.x + threadIdx.x;
  if (i < n) p[i] = 0.0f;
}

// ---------------------------------------------------------------------------
// h_pre = relu(x @ Wh + bh)   (64x512)@(512x512)
// One wave per 16x16 output tile. 4 m-blocks x 32 n-blocks = 128 waves.
// K = 512 in steps of 4 via V_WMMA_F32_16X16X4_F32.
// ---------------------------------------------------------------------------
__global__ void k_gemm_h(const float* __restrict__ x, const float* __restrict__ Wh,
                         const float* __restrict__ bh, float* __restrict__ h_pre) {
  const int lane = threadIdx.x & 31;
  const int wid  = threadIdx.x >> 5;
  const int w    = blockIdx.x * 4 + wid;     // 0..127
  const int m0   = (w >> 5) * 16;            // 0..3  * 16
  const int n0   = (w & 31) * 16;            // 0..31 * 16
  const int half = lane >> 4;                // 0 or 1
  const int r    = lane & 15;

  v8f c = {};
  for (int k = 0; k < 512; k += 4) {
    const int kk = k + 2 * half;             // lanes 0-15: k,k+1 ; lanes 16-31: k+2,k+3
    v2f a, b;
    a.x = x[(m0 + r) * 512 + kk];
    a.y = x[(m0 + r) * 512 + kk + 1];
    b.x = Wh[(size_t)kk * 512 + n0 + r];
    b.y = Wh[(size_t)(kk + 1) * 512 + n0 + r];
    c = __builtin_amdgcn_wmma_f32_16x16x4_f32(false, a, false, b, (short)0, c, false, false);
  }
  const float bias = bh[n0 + r];
#pragma unroll
  for (int j = 0; j < 8; ++j) {
    float v = c[j] + bias;
    v = v > 0.0f ? v : 0.0f;
    h_pre[(size_t)(m0 + j + 8 * half) * 512 + (n0 + r)] = v;
  }
}

// ---------------------------------------------------------------------------
// h = LayerNorm(h_pre) * scale + bias   (per row of 512)
// ---------------------------------------------------------------------------
__global__ void k_ln(const float* __restrict__ h_pre, const float* __restrict__ scale,
                     const float* __restrict__ bias, float* __restrict__ h) {
  __shared__ float s_sum[256];
  __shared__ float s_sq[256];
  const int b = blockIdx.x, t = threadIdx.x;
  const float v0 = h_pre[(size_t)b * 512 + t];
  const float v1 = h_pre[(size_t)b * 512 + 256 + t];
  s_sum[t] = v0 + v1;
  s_sq[t]  = v0 * v0 + v1 * v1;
  __syncthreads();
  for (int off = 128; off > 0; off >>= 1) {
    if (t < off) { s_sum[t] += s_sum[t + off]; s_sq[t] += s_sq[t + off]; }
    __syncthreads();
  }
  const float mean = s_sum[0] * (1.0f / 512.0f);
  const float var  = s_sq[0] * (1.0f / 512.0f) - mean * mean;
  const float rstd = rsqrtf(var + EPSF);
  h[(size_t)b * 512 + t]       = (v0 - mean) * rstd * scale[t]       + bias[t];
  h[(size_t)b * 512 + 256 + t] = (v1 - mean) * rstd * scale[256 + t] + bias[256 + t];
}

// ---------------------------------------------------------------------------
// t1 = x @ Wz1[0:512,:]  and  t2 = x @ Wz2[0:512,:]    (64x512)@(512x4096)
// Exploits repeat structure: z[b*16+a] = t[b] + Wz[512+a].
// 2 matrices x 4 m-blocks x 256 n-blocks = 2048 waves.
// ---------------------------------------------------------------------------
__global__ void k_gemm_t(const float* __restrict__ x, const float* __restrict__ Wz1,
                         const float* __restrict__ Wz2, float* __restrict__ t1,
                         float* __restrict__ t2) {
  const int lane  = threadIdx.x & 31;
  const int wid   = threadIdx.x >> 5;
  const int w     = blockIdx.x * 4 + wid;    // 0..2047
  const int which = w >> 10;                 // uniform per block
  const int rem   = w & 1023;
  const int m0    = (rem >> 8) * 16;         // 4 m-blocks
  const int n0    = (rem & 255) * 16;        // 256 n-blocks
  const float* Wz = which ? Wz2 : Wz1;
  float* t        = which ? t2 : t1;
  const int half  = lane >> 4;
  const int r     = lane & 15;

  v8f c = {};
  for (int k = 0; k < 512; k += 4) {
    const int kk = k + 2 * half;
    v2f a, b;
    a.x = x[(m0 + r) * 512 + kk];
    a.y = x[(m0 + r) * 512 + kk + 1];
    b.x = Wz[(size_t)kk * 4096 + n0 + r];
    b.y = Wz[(size_t)(kk + 1) * 4096 + n0 + r];
    c = __builtin_amdgcn_wmma_f32_16x16x4_f32(false, a, false, b, (short)0, c, false, false);
  }
#pragma unroll
  for (int j = 0; j < 8; ++j)
    t[(size_t)(m0 + j + 8 * half) * 4096 + n0 + r] = c[j];
}

// ---------------------------------------------------------------------------
// Per row r = b*16+a, per 16-g block:
//   w[g,d] = sum_h z2[h,g] * z1[h,d]   (rank-8, two k=4 WMMA per 16x16 tile)
//   normalize over d (mean / (std+eps)), hardware tanh, dot with v=h[b]
//   -> y_[r,g];  atomicAdd(sum_g y^2 / 512) -> q[r]
// Block: 128 threads = 4 waves; each wave computes 8 of the 32 d-tiles.
// Stats/tanh phases use stride-8 LDS access: banks (16g+seg+8i) mod 64 are
// distinct across the 32 lanes of a wave -> conflict-free.
// ---------------------------------------------------------------------------
__global__ void k_bilinear(const float* __restrict__ t1, const float* __restrict__ t2,
                           const float* __restrict__ Wz1, const float* __restrict__ Wz2,
                           const float* __restrict__ h, float* __restrict__ y_,
                           float* __restrict__ q) {
  __shared__ float wbuf[16][528];   // 16 g x 512 d (padded)
  __shared__ float s_v[512];        // v = h[b] staged once
  __shared__ float red[16][8];
  __shared__ float red2[16][8];
  __shared__ float s_mean[16];
  __shared__ float s_rinv[16];
  __shared__ float s_y[16];

  const int gblk = blockIdx.x;      // 0..31
  const int rrow = blockIdx.y;      // 0..1023
  const int b    = rrow >> 4;
  const int a    = rrow & 15;
  const int g0   = gblk * 16;
  const int tid  = threadIdx.x;
  const int lane = tid & 31;
  const int wid  = tid >> 5;
  const int half = lane >> 4;
  const int r    = lane & 15;

  const float* z1t = t1 + (size_t)b * 4096;
  const float* z2t = t2 + (size_t)b * 4096;
  const float* z1w = Wz1 + (size_t)(512 + a) * 4096;
  const float* z2w = Wz2 + (size_t)(512 + a) * 4096;

  // stage v = h[b] into LDS (128 threads x 4 elements)
  const float* vg = h + (size_t)b * 512;
#pragma unroll
  for (int i = 0; i < 4; ++i) s_v[tid + 128 * i] = vg[tid + 128 * i];

  // A fragments: z2 at (h, g0+r).  k-step 0 -> h 0..3, k-step 1 -> h 4..7
  v2f a0, a1;
  {
    const int h0 = 2 * half;
    a0.x = z2t[h0 * 512 + g0 + r]       + z2w[h0 * 512 + g0 + r];
    a0.y = z2t[(h0 + 1) * 512 + g0 + r] + z2w[(h0 + 1) * 512 + g0 + r];
    const int h1 = 4 + 2 * half;
    a1.x = z2t[h1 * 512 + g0 + r]       + z2w[h1 * 512 + g0 + r];
    a1.y = z2t[(h1 + 1) * 512 + g0 + r] + z2w[(h1 + 1) * 512 + g0 + r];
  }

  for (int i = 0; i < 8; ++i) {
    const int d0 = (wid * 8 + i) * 16;
    const int h0 = 2 * half;
    const int h1 = 4 + 2 * half;
    v2f b0, b1;
    b0.x = z1t[h0 * 512 + d0 + r]       + z1w[h0 * 512 + d0 + r];
    b0.y = z1t[(h0 + 1) * 512 + d0 + r] + z1w[(h0 + 1) * 512 + d0 + r];
    b1.x = z1t[h1 * 512 + d0 + r]       + z1w[h1 * 512 + d0 + r];
    b1.y = z1t[(h1 + 1) * 512 + d0 + r] + z1w[(h1 + 1) * 512 + d0 + r];
    v8f c = {};
    c = __builtin_amdgcn_wmma_f32_16x16x4_f32(false, a0, false, b0, (short)0, c, false, false);
    c = __builtin_amdgcn_wmma_f32_16x16x4_f32(false, a1, false, b1, (short)0, c, false, false);
#pragma unroll
    for (int j = 0; j < 8; ++j)
      wbuf[j + 8 * half][d0 + r] = c[j];
  }
  __syncthreads();

  // Row stats over d=0..511 for each of 16 g's: 8 threads per g, stride 8.
  const int g   = tid >> 3;   // 0..15
  const int seg = tid & 7;    // 0..7
  float s = 0.0f, sq = 0.0f;
  for (int d = seg; d < 512; d += 8) {
    const float wv = wbuf[g][d];
    s += wv;
    sq += wv * wv;
  }
  red[g][seg]  = s;
  red2[g][seg] = sq;
  __syncthreads();
  if (seg == 0) {
    float S = 0.0f, SQ = 0.0f;
    for (int k = 0; k < 8; ++k) { S += red[g][k]; SQ += red2[g][k]; }
    const float mean = S * (1.0f / 512.0f);
    float var = SQ * (1.0f / 512.0f) - mean * mean;
    var = var > 0.0f ? var : 0.0f;
    s_mean[g] = mean;
    s_rinv[g] = 1.0f / (sqrtf(var) + EPSF);   // matches (w-mu)/(std+eps)
  }
  __syncthreads();

  // y_[r, g0+g] = sum_d tanh(w*rinv - mu*rinv) * v[d]
  const float rinv = s_rinv[g];
  const float nmu  = -s_mean[g] * rinv;
  float acc = 0.0f;
  for (int d = seg; d < 512; d += 8)
    acc += fast_tanh(fmaf(wbuf[g][d], rinv, nmu)) * s_v[d];
  red[g][seg] = acc;
  __syncthreads();
  if (seg == 0) {
    float Y = 0.0f;
    for (int k = 0; k < 8; ++k) Y += red[g][k];
    y_[(size_t)rrow * 512 + g0 + g] = Y;
    s_y[g] = Y;
  }
  __syncthreads();
  if (tid == 0) {
    float qs = 0.0f;
    for (int k = 0; k < 16; ++k) qs += s_y[k] * s_y[k];
    atomicAdd(&q[rrow], qs * (1.0f / 512.0f));
  }
}

// ---------------------------------------------------------------------------
extern "C" void kernel_launch(void* const* d_in, const int* in_sizes, int n_in,
                              void* d_out, int out_size, void* d_ws, size_t ws_size,
                              hipStream_t stream) {
  const float* x        = (const float*)d_in[0];  // (64,512)
  const float* Wh       = (const float*)d_in[1];  // (512,512)
  const float* bh       = (const float*)d_in[2];  // (512,)
  const float* ln_scale = (const float*)d_in[3];  // (512,)
  const float* ln_bias  = (const float*)d_in[4];  // (512,)
  const float* Wz1      = (const float*)d_in[5];  // (528,4096)
  const float* Wz2      = (const float*)d_in[6];  // (528,4096)

  float* out   = (float*)d_out;
  float* h_out = out;                 // 64*512  = 32768
  float* q_out = out + 64 * 512;      // 64*16   = 1024
  float* y_out = q_out + 64 * 16;     // 1024*512= 524288

  float* ws    = (float*)d_ws;
  float* h_pre = ws;                  // 32768 floats
  float* t1    = ws + 32768;          // 262144 floats
  float* t2    = t1 + 262144;         // 262144 floats

  k_zero<<<4, 256, 0, stream>>>(q_out, 1024);
  k_gemm_h<<<32, 128, 0, stream>>>(x, Wh, bh, h_pre);
  k_ln<<<64, 256, 0, stream>>>(h_pre, ln_scale, ln_bias, h_out);
  k_gemm_t<<<512, 128, 0, stream>>>(x, Wz1, Wz2, t1, t2);
  dim3 g3(32, 1024);
  k_bilinear<<<g3, 128, 0, stream>>>(t1, t2, Wz1, Wz2, h_out, y_out, q_out);
}